// MeanStdFilter_28235115004165
// MI455X (gfx1250) — compile-verified
//
#include <hip/hip_runtime.h>
#include <math.h>

typedef float v4f __attribute__((ext_vector_type(4)));
typedef float v2f __attribute__((ext_vector_type(2)));
typedef float v8f __attribute__((ext_vector_type(8)));

// ---------------------------------------------------------------------------
// Pass 1: per-chunk partial column sums / sums of squares.
// Thread t owns float4 column-group c4; loops over the rows of its chunk.
// Wave access: 32 lanes * 16B = 512B contiguous per row -> fully coalesced.
// ---------------------------------------------------------------------------
__global__ void msf_partial(const float* __restrict__ x,
                            float* __restrict__ psum,
                            float* __restrict__ psumsq,
                            int F, int rowsPerChunk) {
    const int F4 = F >> 2;
    const int c4 = blockIdx.x * blockDim.x + threadIdx.x;
    if (c4 >= F4) return;
    const int chunk = blockIdx.y;
    const v4f* __restrict__ xv = (const v4f*)x;

    long long base = (long long)chunk * rowsPerChunk * F4 + c4;
    v4f s = {0.f, 0.f, 0.f, 0.f};
    v4f q = {0.f, 0.f, 0.f, 0.f};
    for (int r = 0; r < rowsPerChunk; ++r) {
        v4f v = xv[base + (long long)r * F4];
        s += v;
        q += v * v;   // fused to v_fma by the compiler
    }
    ((v4f*)psum)  [(size_t)chunk * F4 + c4] = s;
    ((v4f*)psumsq)[(size_t)chunk * F4 + c4] = q;
}

// ---------------------------------------------------------------------------
// Pass 2: reduce partials across chunks with V_WMMA_F32_16X16X4_F32.
// A = all-ones 16x4  =>  D[m][n] = sum_k B[k][n] + C[m][n]  (every row of D
// identical, result independent of the exact K<->lane assignment of B).
// One wave per 16 columns; B tile lanes: N = lane%16, K-pair = 2*(lane/16).
// Then finalize mean / 1/(std+eps) per column.
// ---------------------------------------------------------------------------
__global__ void msf_finalize(const float* __restrict__ psum,
                             const float* __restrict__ psumsq,
                             const float* __restrict__ S0,
                             float* __restrict__ mean,
                             float* __restrict__ inv,
                             int F, int chunks, float Bn, float eps) {
    const int lane    = threadIdx.x;        // wave32
    const int colBase = blockIdx.x * 16;
    const int n       = lane & 15;
    const int kOff    = (lane >> 4) * 2;    // 0 or 2

    v2f a;  a.x = 1.0f; a.y = 1.0f;         // all-ones A (16x4)
    v8f c0 = {};                            // sum accumulator
    v8f c1 = {};                            // sumsq accumulator

    for (int k = 0; k < chunks; k += 4) {
        v2f b0, b1;
        b0.x = psum  [(size_t)(k + kOff)     * F + colBase + n];
        b0.y = psum  [(size_t)(k + kOff + 1) * F + colBase + n];
        b1.x = psumsq[(size_t)(k + kOff)     * F + colBase + n];
        b1.y = psumsq[(size_t)(k + kOff + 1) * F + colBase + n];
        c0 = __builtin_amdgcn_wmma_f32_16x16x4_f32(false, a, false, b0,
                                                   (short)0, c0, false, false);
        c1 = __builtin_amdgcn_wmma_f32_16x16x4_f32(false, a, false, b1,
                                                   (short)0, c1, false, false);
    }

    if (lane < 16) {                        // after all WMMAs (EXEC all-1s above)
        float sum   = c0[0];                // any D row: column sum for N = n
        float sumsq = c1[0];
        float m     = sum / Bn;
        float S     = S0[colBase + n] + (sumsq - Bn * m * m);
        float var   = S / (Bn - 1.0f);
        float sd    = sqrtf(var);
        mean[colBase + n] = m;
        inv [colBase + n] = 1.0f / (sd + eps);
    }
}

// ---------------------------------------------------------------------------
// Pass 3: out = (x - mean) * inv.  x re-read (L2-resident), output written
// with non-temporal b128 stores so the write-once stream doesn't evict x.
// mean/inv loaded once per thread, reused across the row chunk.
// ---------------------------------------------------------------------------
__global__ void msf_norm(const float* __restrict__ x,
                         const float* __restrict__ mean,
                         const float* __restrict__ inv,
                         float* __restrict__ out,
                         int F, int rowsPerChunk) {
    const int F4 = F >> 2;
    const int c4 = blockIdx.x * blockDim.x + threadIdx.x;
    if (c4 >= F4) return;

    const v4f m  = ((const v4f*)mean)[c4];
    const v4f iv = ((const v4f*)inv)[c4];
    const v4f* __restrict__ xv = (const v4f*)x;
    v4f* __restrict__ ov = (v4f*)out;

    long long base = (long long)blockIdx.y * rowsPerChunk * F4 + c4;
    for (int r = 0; r < rowsPerChunk; ++r) {
        long long idx = base + (long long)r * F4;
        v4f v = xv[idx];
        v4f o = (v - m) * iv;
        __builtin_nontemporal_store(o, &ov[idx]);
    }
}

// ---------------------------------------------------------------------------
extern "C" void kernel_launch(void* const* d_in, const int* in_sizes, int n_in,
                              void* d_out, int out_size, void* d_ws, size_t ws_size,
                              hipStream_t stream) {
    const float* x  = (const float*)d_in[0];
    const float* S0 = (const float*)d_in[2];   // running M2 buffer (zeros at init)
    float* out = (float*)d_out;

    const int F  = in_sizes[1];                // 1024
    const int Bn = in_sizes[0] / F;            // 32768
    const int F4 = F >> 2;

    float* ws   = (float*)d_ws;
    float* mean = ws;
    float* inv  = ws + F;

    // Pick chunk count: power of two, >=4, divides Bn, partials fit in ws.
    int chunks = 4;
    while (chunks < 512) {
        int nx = chunks * 2;
        size_t need = (size_t)(2 + 2 * (size_t)nx) * F * sizeof(float);
        if ((Bn % nx) == 0 && need <= ws_size) chunks = nx; else break;
    }
    float* psum   = ws + 2 * F;
    float* psumsq = psum + (size_t)chunks * F;

    const int tpb = 256;
    dim3 g1((F4 + tpb - 1) / tpb, chunks);
    msf_partial<<<g1, tpb, 0, stream>>>(x, psum, psumsq, F, Bn / chunks);

    msf_finalize<<<F / 16, 32, 0, stream>>>(psum, psumsq, S0, mean, inv,
                                            F, chunks, (float)Bn, 1e-5f);

    int chunks3 = 512;                         // parallelism for the normalize pass
    while (chunks3 > 1 && (Bn % chunks3) != 0) chunks3 >>= 1;
    dim3 g3((F4 + tpb - 1) / tpb, chunks3);
    msf_norm<<<g3, tpb, 0, stream>>>(x, mean, inv, out, F, Bn / chunks3);
}